// SuperLoss2_18983755448980
// MI455X (gfx1250) — compile-verified
//
#include <hip/hip_runtime.h>
#include <cstdint>

typedef float v2f __attribute__((ext_vector_type(2)));
typedef float v8f __attribute__((ext_vector_type(8)));

#define BLOCK       256
#define NBLOCKS     1024
#define TILE_ROWS   1024
#define TILE_BYTES  (TILE_ROWS * 12)      // 12288 B per tile (1024 rows x 3 f32)
#define FILLS       3                      // 3 x (256 lanes * 16B) = 12288 B
#define HALLEY_ITERS 8

#define TAUC 0.6931471805599453f           // log(2)
#define INV_LAM 4.0f                       // 1/0.25
#define NEG_2_OVER_E (-0.7357588823428847f)
#define E_CONST 2.718281828459045f

// Raw v_rcp_f32-based division: ~2 VALU ops instead of the ~11-op IEEE expansion.
// Safe here: callers guard the only denominator that can reach zero.
__device__ __forceinline__ float fast_div(float a, float b) {
  return a * __builtin_amdgcn_rcpf(b);
}

__device__ __forceinline__ void async_fill_tile(uint32_t ldsBase, const char* gbase,
                                                long long tileByte, long long totalBytes,
                                                int tid) {
#pragma unroll
  for (int k = 0; k < FILLS; ++k) {
    long long off = tileByte + (long long)(k * (BLOCK * 16) + tid * 16);
    if (off + 16 <= totalBytes) {
      uint32_t lds  = ldsBase + (uint32_t)(k * (BLOCK * 16) + tid * 16);
      uint32_t voff = (uint32_t)off;   // 75.5 MB input: fits 32-bit GVS voffset
      asm volatile("global_load_async_to_lds_b128 %0, %1, %2"
                   :: "v"(lds), "v"(voff), "s"(gbase) : "memory");
    }
  }
}

__device__ __forceinline__ float superloss_row(float x0, float x1, float x2, int lab) {
  // cross entropy for 3 classes
  float xl = (lab == 0) ? x0 : ((lab == 1) ? x1 : x2);
  float m  = fmaxf(x0, fmaxf(x1, x2));
  float s  = __expf(x0 - m) + __expf(x1 - m) + __expf(x2 - m);
  float l  = m + __logf(s) - xl;

  // Lambert W (principal branch), Halley iteration (cubic convergence:
  // 8 iterations is ample margin from this branch-point-aware init).
  float y = 0.5f * fmaxf(NEG_2_OVER_E, (l - TAUC) * INV_LAM);
  float w = (y < 1.0f)
          ? (-1.0f + __builtin_amdgcn_sqrtf(fmaxf(2.0f * fmaf(E_CONST, y, 1.0f), 0.0f)))
          : __logf(1.0f + y);
#pragma unroll
  for (int it = 0; it < HALLEY_ITERS; ++it) {
    float ew    = __expf(w);
    float f     = fmaf(w, ew, -y);
    float t     = fast_div((w + 2.0f) * f, fmaf(2.0f, w, 2.0f) + 1e-12f);
    float denom = fmaf(ew, w + 1.0f, -t);
    // guard: at the branch point (w=-1, f=0) denom can be 0 -> 0*inf NaN
    w = (fabsf(f) > 0.0f) ? (w - fast_div(f, denom)) : w;
  }
  float sigma = __expf(-w);
  return fmaf(l - TAUC, sigma, 0.25f * w * w);
}

__global__ void __launch_bounds__(BLOCK)
superloss_partial(const float* __restrict__ logits,
                  const int* __restrict__ n0p,
                  const int* __restrict__ n1p,
                  float* __restrict__ partials,
                  int Nrows, int tilesPerBlock, int tilesTotal) {
  __shared__ __align__(16) float buf[2][TILE_BYTES / 4];
  __shared__ float red[BLOCK];

  const int tid = threadIdx.x;
  const long long totalBytes = (long long)Nrows * 12;
  const char* gbase = (const char*)logits;

  const int n0  = n0p[0];
  const int n01 = n0 + n1p[0];

  int tile0   = blockIdx.x * tilesPerBlock;
  int myTiles = tilesTotal - tile0;
  if (myTiles > tilesPerBlock) myTiles = tilesPerBlock;
  if (myTiles < 0) myTiles = 0;

  float acc = 0.0f;
  if (myTiles > 0) {
    uint32_t lds0 = (uint32_t)(uintptr_t)&buf[0][0];
    uint32_t lds1 = (uint32_t)(uintptr_t)&buf[1][0];

    // prologue: fill tile 0
    async_fill_tile(lds0, gbase, (long long)tile0 * TILE_BYTES, totalBytes, tid);

    for (int t = 0; t < myTiles; ++t) {
      if (t + 1 < myTiles) {
        // prefetch next tile into the other buffer, then wait for current tile
        async_fill_tile((t & 1) ? lds0 : lds1, gbase,
                        (long long)(tile0 + t + 1) * TILE_BYTES, totalBytes, tid);
        asm volatile("s_wait_asynccnt 0x3" ::: "memory");
      } else {
        asm volatile("s_wait_asynccnt 0x0" ::: "memory");
      }
      __syncthreads();   // all waves' fills for tile t have landed

      const float* tb = &buf[t & 1][0];
      const int rowBase = (tile0 + t) * TILE_ROWS;   // Nrows < 2^31: 32-bit math
#pragma unroll
      for (int j = 0; j < 4; ++j) {
        int slot = j * BLOCK + tid;          // lane-stride-3 dwords: conflict-light
        int r = rowBase + slot;
        if (r < Nrows) {
          float x0 = tb[slot * 3 + 0];
          float x1 = tb[slot * 3 + 1];
          float x2 = tb[slot * 3 + 2];
          int lab = (int)(r >= n0) + (int)(r >= n01);
          acc += superloss_row(x0, x1, x2, lab);
        }
      }
      __syncthreads();   // buffer free for the prefetch issued next iteration
    }
  }

  // ---- block reduction: 256 partials summed on the matrix engine ----
  red[tid] = acc;
  __syncthreads();
  if (tid < 32) {
    v2f a; a.x = 1.0f; a.y = 1.0f;          // all-ones A => D rows = column sums of B
    v8f c = {0.f, 0.f, 0.f, 0.f, 0.f, 0.f, 0.f, 0.f};
#pragma unroll
    for (int jj = 0; jj < 4; ++jj) {
      v2f b;
      b.x = red[jj * 64 + tid];
      b.y = red[jj * 64 + 32 + tid];
      c = __builtin_amdgcn_wmma_f32_16x16x4_f32(false, a, false, b, (short)0, c,
                                                false, false);
    }
    // c[0] across 32 lanes holds each column sum twice -> wave-sum / 2 = total
    float v = c[0];
#pragma unroll
    for (int o = 16; o > 0; o >>= 1) v += __shfl_xor(v, o, 32);
    if (tid == 0) partials[blockIdx.x] = 0.5f * v;
  }
}

__global__ void __launch_bounds__(BLOCK)
superloss_finalize(const float* __restrict__ partials, float* __restrict__ out, int n) {
  __shared__ float red[BLOCK];
  float acc = 0.0f;
  for (int i = threadIdx.x; i < n; i += BLOCK) acc += partials[i];
  red[threadIdx.x] = acc;
  __syncthreads();
  for (int s = BLOCK / 2; s > 0; s >>= 1) {
    if (threadIdx.x < s) red[threadIdx.x] += red[threadIdx.x + s];
    __syncthreads();
  }
  if (threadIdx.x == 0) out[0] = red[0] * (1.0f / 12.0f);   // / BATCH_SIZE
}

extern "C" void kernel_launch(void* const* d_in, const int* in_sizes, int n_in,
                              void* d_out, int out_size, void* d_ws, size_t ws_size,
                              hipStream_t stream) {
  const float* logits = (const float*)d_in[0];
  const int*   n0p    = (const int*)d_in[1];
  const int*   n1p    = (const int*)d_in[2];
  // d_in[3] (n2) unused: label 2 is "everything past n0+n1"

  int Nrows         = in_sizes[0] / 3;
  int tilesTotal    = (Nrows + TILE_ROWS - 1) / TILE_ROWS;
  int tilesPerBlock = (tilesTotal + NBLOCKS - 1) / NBLOCKS;

  float* partials = (float*)d_ws;  // NBLOCKS floats = 4 KB scratch

  superloss_partial<<<NBLOCKS, BLOCK, 0, stream>>>(logits, n0p, n1p, partials,
                                                   Nrows, tilesPerBlock, tilesTotal);
  superloss_finalize<<<1, BLOCK, 0, stream>>>(partials, (float*)d_out, NBLOCKS);
}